// TinyRNN_13872744366607
// MI455X (gfx1250) — compile-verified
//
#include <hip/hip_runtime.h>
#include <hip/hip_bf16.h>
#include <math.h>

// TinyRNN on MI455X (gfx1250, wave32):
//   h_{t+1}[j,b] = tanh( x[b,t]*W_ih[j] + b_ih[j] + b_hh[j] + sum_k W_hh[j,k]*h_t[k,b] )
//
// One wave owns a 16-sequence batch tile. K is padded 12 -> 16 and the K axis
// is RELABELED so that the WMMA D/C register layout (j = v + 8*half) coincides
// exactly with the B-operand layout (physical slot kp = r + 2*half per slice):
//   slot (slice s, reg r, half h)  <- hidden index j = 2s + r + 8h
// With that permutation, B slice s == {D[2s], D[2s+1]} verbatim: the tanh
// outputs feed the next step's WMMAs with ZERO cross-lane data movement.
// A (W_hh) is preloaded once with the matching column permutation, zero-padded
// for labels >= 12 and rows >= 12. Per step: 8 FMA + 4 chained
// V_WMMA_F32_16X16X4_F32 + 8 v_tanh_f32 — nothing else on the serial chain.

typedef float v2f __attribute__((ext_vector_type(2)));
typedef float v8f __attribute__((ext_vector_type(8)));

#define HDIM 12
#define ODIM 2
#define WAVES_PER_BLOCK 4

__device__ __forceinline__ float fast_tanhf(float x) {
#if __has_builtin(__builtin_amdgcn_tanhf)
  return __builtin_amdgcn_tanhf(x);
#elif __has_builtin(__builtin_amdgcn_tanh_f32)
  return __builtin_amdgcn_tanh_f32(x);
#else
  return tanhf(x);
#endif
}

__launch_bounds__(32 * WAVES_PER_BLOCK)
__global__ void tinyrnn_wmma_kernel(const float* __restrict__ x,
                                    const float* __restrict__ W_ih,
                                    const float* __restrict__ W_hh,
                                    const float* __restrict__ b_ih,
                                    const float* __restrict__ b_hh,
                                    const float* __restrict__ W_head,
                                    const float* __restrict__ b_head,
                                    float* __restrict__ out,
                                    int T, int B) {
  const int lane = threadIdx.x & 31;
  const int wv   = threadIdx.x >> 5;                          // wave in block
  const int base = (blockIdx.x * WAVES_PER_BLOCK + wv) * 16;  // batch tile base

  const int col  = lane & 15;    // batch column within tile (N index)
  const int half = lane >> 4;    // 0: lanes 0-15, 1: lanes 16-31

  // ---- A operand: W_hh with permuted K labels, 4 slices x v2f.
  // A reg (slice s, r), lane: row M = lane%16, label j = 2s + r + 8*half.
  float a[4][2];
#pragma unroll
  for (int s = 0; s < 4; ++s) {
#pragma unroll
    for (int r = 0; r < 2; ++r) {
      const int j = 2 * s + r + 8 * half;   // K label carried by this slot
      a[s][r] = (col < HDIM && j < HDIM) ? W_hh[col * HDIM + j] : 0.0f;
    }
  }

  // ---- Per-row constants in C/D layout: reg v, lane -> j = v + 8*half.
  float wih[8], bias[8];
#pragma unroll
  for (int v = 0; v < 8; ++v) {
    const int j = v + 8 * half;
    wih[v]  = (j < HDIM) ? W_ih[j] : 0.0f;
    bias[v] = (j < HDIM) ? (b_ih[j] + b_hh[j]) : 0.0f;
  }

  // ---- x stream: lane reads row (base+col); lanes 16-31 duplicate 0-15.
  int brow = base + col;
  if (brow >= B) brow = B - 1;
  const float4* __restrict__ xrow =
      (const float4*)(x + (size_t)brow * (size_t)T);

  // h lives in D/C layout registers and doubles as the B operand. h0 = 0,
  // and rows 12..15 stay exactly 0 (zero A rows + zero bias), so the padded
  // K slots always contribute 0.
  float d[8];
#pragma unroll
  for (int i = 0; i < 8; ++i) d[i] = 0.0f;

  auto step = [&](float xt) {
    v8f c;
#pragma unroll
    for (int v = 0; v < 8; ++v) c[v] = fmaf(xt, wih[v], bias[v]);
    const v2f A0 = {a[0][0], a[0][1]};
    const v2f A1 = {a[1][0], a[1][1]};
    const v2f A2 = {a[2][0], a[2][1]};
    const v2f A3 = {a[3][0], a[3][1]};
    const v2f B0 = {d[0], d[1]};   // labels j = {0,1} lo / {8,9}  hi
    const v2f B1 = {d[2], d[3]};   // labels j = {2,3} lo / {10,11} hi
    const v2f B2 = {d[4], d[5]};   // labels j = {4,5} lo / {12,13}=0
    const v2f B3 = {d[6], d[7]};   // labels j = {6,7} lo / {14,15}=0
    c = __builtin_amdgcn_wmma_f32_16x16x4_f32(false, A0, false, B0,
                                              (short)0, c, false, false);
    c = __builtin_amdgcn_wmma_f32_16x16x4_f32(false, A1, false, B1,
                                              (short)0, c, false, false);
    c = __builtin_amdgcn_wmma_f32_16x16x4_f32(false, A2, false, B2,
                                              (short)0, c, false, false);
    c = __builtin_amdgcn_wmma_f32_16x16x4_f32(false, A3, false, B3,
                                              (short)0, c, false, false);
#pragma unroll
    for (int v = 0; v < 8; ++v) d[v] = fast_tanhf(c[v]);
  };

  // ---- serial scan over T, float4-batched x with one-ahead prefetch
  const int NT4 = T >> 2;
  float4 xv = xrow[0];
  for (int tq = 0; tq < NT4; ++tq) {
    float4 xn = xv;
    if (tq + 1 < NT4) xn = xrow[tq + 1];
    step(xv.x);
    step(xv.y);
    step(xv.z);
    step(xv.w);
    xv = xn;
  }
  const float* xs = x + (size_t)brow * (size_t)T;
  for (int t = NT4 << 2; t < T; ++t) step(xs[t]);  // tail (T%4 == 0 for 2048)

  // ---- head: out[b,o] = b_head[o] + sum_j h[j,b] * W_head[o*H + j]
  __shared__ float hbuf[WAVES_PER_BLOCK][16][HDIM];
#pragma unroll
  for (int v = 0; v < 8; ++v) {
    const int j = v + 8 * half;
    if (j < HDIM) hbuf[wv][col][j] = d[v];
  }
  __syncthreads();
  const int r = col, o = half;   // lane -> (row, head output)
  const int ob = base + r;
  if (ob < B) {
    float acc = b_head[o];
#pragma unroll
    for (int j = 0; j < HDIM; ++j)
      acc = fmaf(hbuf[wv][r][j], W_head[o * HDIM + j], acc);
    out[ob * ODIM + o] = acc;
  }
}

extern "C" void kernel_launch(void* const* d_in, const int* in_sizes, int n_in,
                              void* d_out, int out_size, void* d_ws, size_t ws_size,
                              hipStream_t stream) {
  (void)n_in; (void)d_ws; (void)ws_size;
  const float* x      = (const float*)d_in[0];
  const float* W_ih   = (const float*)d_in[1];
  const float* W_hh   = (const float*)d_in[2];
  const float* b_ih   = (const float*)d_in[3];
  const float* b_hh   = (const float*)d_in[4];
  const float* W_head = (const float*)d_in[5];
  const float* b_head = (const float*)d_in[6];
  float* out = (float*)d_out;

  const int B = out_size / ODIM;          // 4096
  const int T = in_sizes[0] / B;          // 2048 (I == 1)

  const int waves  = (B + 15) / 16;                        // 256 wave-chains
  const int blocks = (waves + WAVES_PER_BLOCK - 1) / WAVES_PER_BLOCK;
  tinyrnn_wmma_kernel<<<blocks, 32 * WAVES_PER_BLOCK, 0, stream>>>(
      x, W_ih, W_hh, b_ih, b_hh, W_head, b_head, out, T, B);
}